// NGCFConv_78168404787215
// MI455X (gfx1250) — compile-verified
//
#include <hip/hip_runtime.h>
#include <hip/hip_bf16.h>

typedef __attribute__((ext_vector_type(2))) float v2f;
typedef __attribute__((ext_vector_type(8))) float v8f;

#define D 128

// ---------------- Kernel 1: zero the aggregation buffer (workspace) ----------
__global__ __launch_bounds__(256) void ngcf_zero(float4* __restrict__ agg, int n4) {
    int i = blockIdx.x * 256 + threadIdx.x;
    if (i < n4) agg[i] = make_float4(0.f, 0.f, 0.f, 0.f);
}

// ---------------- Kernel 2: edge gather-scale-scatter ------------------------
// One wave (32 lanes) per edge; lane handles 4 consecutive features.
__global__ __launch_bounds__(256) void ngcf_scatter(
    const int* __restrict__ ei,      // [2, E] (row ptrs, then col ptrs)
    const float* __restrict__ ew,    // [E]
    const float* __restrict__ x,     // [N, 128]
    float* __restrict__ agg,         // [N, 128]
    int E)
{
    const int wave = threadIdx.x >> 5;
    const int lane = threadIdx.x & 31;
    const int e = blockIdx.x * 8 + wave;
    if (e >= E) return;

    const int   src = ei[e];        // wave-uniform -> scalar loads
    const int   dst = ei[E + e];
    const float w   = ew[e];

    const float4 xv = *(const float4*)(x + src * D + lane * 4);
    float* p = agg + dst * D + lane * 4;
    atomicAdd(p + 0, w * xv.x);
    atomicAdd(p + 1, w * xv.y);
    atomicAdd(p + 2, w * xv.z);
    atomicAdd(p + 3, w * xv.w);
}

// ---------------- Kernel 3: fused dual-GEMM + bias + leaky ReLU --------------
// Block = 256 threads = 8 waves; handles a 16-row node tile.
// Wave j computes the 16x16 output tile at columns [16j, 16j+16) using
// V_WMMA_F32_16X16X4_F32, accumulating  (agg+x)@W1^T  and  (agg*x)@W2^T.
__global__ __launch_bounds__(256) void ngcf_gemm(
    const float* __restrict__ x,
    const float* __restrict__ agg,
    const float* __restrict__ W1, const float* __restrict__ b1,
    const float* __restrict__ W2, const float* __restrict__ b2,
    float* __restrict__ out)
{
    __shared__ float sSum[16 * D];   // agg + x   tile (16 rows)
    __shared__ float sMul[16 * D];   // agg * x   tile

    const int itile = blockIdx.x;
    const int base  = itile * 16 * D;

    // Coalesced cooperative stage of the A-tiles into LDS.
    #pragma unroll
    for (int idx = threadIdx.x; idx < 16 * D; idx += 256) {
        const float xv = x[base + idx];
        const float av = agg[base + idx];
        sSum[idx] = av + xv;
        sMul[idx] = av * xv;
    }
    __syncthreads();

    const int wave  = threadIdx.x >> 5;
    const int lane  = threadIdx.x & 31;
    const int half  = lane >> 4;          // 0: K pair {k,k+1}, 1: {k+2,k+3}
    const int l     = lane & 15;
    const int jtile = wave;
    const int bcol  = jtile * 16 + l;     // this lane's B column (= output col)

    // A fragment source: row = l, K offset = 2*half within each K-step of 4.
    const int aoff = l * D + 2 * half;
    const float* wr1 = W1 + bcol * D + 2 * half;   // W row = B column (B = W^T)
    const float* wr2 = W2 + bcol * D + 2 * half;

    v8f c1 = {};   // (agg+x) @ W1^T accumulator
    v8f c2 = {};   // (agg*x) @ W2^T accumulator

    #pragma unroll 4
    for (int k = 0; k < D; k += 4) {
        const v2f aS = *(const v2f*)(sSum + aoff + k);
        const v2f aM = *(const v2f*)(sMul + aoff + k);
        const v2f bA = *(const v2f*)(wr1 + k);
        const v2f bB = *(const v2f*)(wr2 + k);
        c1 = __builtin_amdgcn_wmma_f32_16x16x4_f32(
                 false, aS, false, bA, (short)0, c1, false, false);
        c2 = __builtin_amdgcn_wmma_f32_16x16x4_f32(
                 false, aM, false, bB, (short)0, c2, false, false);
    }

    const float bias = b1[bcol] + b2[bcol];

    // C/D layout: lane (l, half) holds col = l, rows = v + 8*half, v = 0..7.
    #pragma unroll
    for (int v = 0; v < 8; ++v) {
        const int row = itile * 16 + v + 8 * half;
        const float h = c1[v] + c2[v] + bias;
        out[row * D + bcol] = fmaxf(h, 0.2f * h);   // leaky ReLU, slope 0.2
    }
}

// ---------------- Launcher ---------------------------------------------------
extern "C" void kernel_launch(void* const* d_in, const int* in_sizes, int n_in,
                              void* d_out, int out_size, void* d_ws, size_t ws_size,
                              hipStream_t stream) {
    const float* x  = (const float*)d_in[0];
    const int*   ei = (const int*)  d_in[1];
    const float* ew = (const float*)d_in[2];
    const float* W1 = (const float*)d_in[3];
    const float* b1 = (const float*)d_in[4];
    const float* W2 = (const float*)d_in[5];
    const float* b2 = (const float*)d_in[6];
    float* out = (float*)d_out;
    float* agg = (float*)d_ws;                    // N*D floats (51.2 MB)

    const int N = in_sizes[0] / D;                // 100000
    const int E = in_sizes[2];                    // 600000

    const int n4 = (N * D) / 4;
    ngcf_zero<<<(n4 + 255) / 256, 256, 0, stream>>>((float4*)agg, n4);
    ngcf_scatter<<<(E + 7) / 8, 256, 0, stream>>>(ei, ew, x, agg, E);
    ngcf_gemm<<<N / 16, 256, 0, stream>>>(x, agg, W1, b1, W2, b2, out);
}